// Attention_21698174780065
// MI455X (gfx1250) — compile-verified
//
#include <hip/hip_runtime.h>
#include <math.h>

typedef __attribute__((ext_vector_type(2))) float v2f;
typedef __attribute__((ext_vector_type(8))) float v8f;

#define GN 16384            // spatial pixels per (b, channel) plane: 128*128
#define DIM 192
#define HEADS 4
#define HDIM 48
#define QKV_CH 576          // 3*DIM
#define NSPLIT 32           // K-splits for the Gram matrix (parallelism)

// ---------------------------------------------------------------------------
// Generic WMMA f32 GEMM:  Y[b, m, n] = sum_k W[b][m,k] * X[b, chanOff+k, n]
// W row-major [M,K<=192] (per-batch stride wBatchStride, 0 = shared weights).
// Block = 256 threads = 8 waves; wave computes a 16(M) x 64(N) tile.
// W tile (16 x K) is staged once in padded LDS: A-frags come from ds_load,
// B-frags stream from global (L2-resident per n-tile).
// Grid = (M/16, GN/512, B).
// ---------------------------------------------------------------------------
__global__ void gemm_wmma(const float* __restrict__ W, long wBatchStride,
                          const float* __restrict__ X, long xBatchStride, int chanOff,
                          float* __restrict__ Y, int M, int K) {
  __shared__ float Ws[16][193];             // 193 % 64 == 1 -> conflict-free
  const int b    = blockIdx.z;
  const int m0   = blockIdx.x * 16;
  const int lane = threadIdx.x & 31;
  const int wave = threadIdx.x >> 5;
  const int n0   = blockIdx.y * 512 + wave * 64;
  const int l16  = lane & 15;
  const int kk   = (lane < 16) ? 0 : 2;     // A/B frag K-pair per lane half

  // Stage the block's 16 x K weight tile into LDS (coalesced along K).
  const float* Wp = W + (size_t)b * wBatchStride + (size_t)m0 * K;
  for (int i = threadIdx.x; i < 16 * K; i += 256) {
    int r = i / K, c = i - r * K;
    Ws[r][c] = Wp[(size_t)r * K + c];
  }
  __syncthreads();

  const float* Xp = X + (size_t)b * xBatchStride + (size_t)chanOff * GN;

  v8f acc0 = {}, acc1 = {}, acc2 = {}, acc3 = {};

  for (int k0 = 0; k0 < K; k0 += 4) {
    // A fragment from LDS: 16x4 f32 -> 2 VGPRs per lane
    v2f a;
    a.x = Ws[l16][k0 + kk];
    a.y = Ws[l16][k0 + kk + 1];

    const float* xr0 = Xp + (size_t)(k0 + kk) * GN + n0 + l16;
    const float* xr1 = xr0 + GN;
    __builtin_prefetch(xr0 + 4 * GN, 0, 1);   // global_prefetch_b8 for next K rows

    v2f b0; b0.x = xr0[0];  b0.y = xr1[0];
    v2f b1; b1.x = xr0[16]; b1.y = xr1[16];
    v2f b2; b2.x = xr0[32]; b2.y = xr1[32];
    v2f b3; b3.x = xr0[48]; b3.y = xr1[48];

    acc0 = __builtin_amdgcn_wmma_f32_16x16x4_f32(false, a, false, b0, (short)0, acc0, false, false);
    acc1 = __builtin_amdgcn_wmma_f32_16x16x4_f32(false, a, false, b1, (short)0, acc1, false, false);
    acc2 = __builtin_amdgcn_wmma_f32_16x16x4_f32(false, a, false, b2, (short)0, acc2, false, false);
    acc3 = __builtin_amdgcn_wmma_f32_16x16x4_f32(false, a, false, b3, (short)0, acc3, false, false);
  }

  // C/D layout: VGPR r -> M = m0 + r (+8 for lanes 16..31), N = n-base + lane%16
  const int mst = m0 + ((lane < 16) ? 0 : 8);
#pragma unroll
  for (int r = 0; r < 8; ++r) {
    size_t base = ((size_t)b * M + (mst + r)) * GN + n0 + l16;
    Y[base]      = acc0[r];
    Y[base + 16] = acc1[r];
    Y[base + 32] = acc2[r];
    Y[base + 48] = acc3[r];
  }
}

// ---------------------------------------------------------------------------
// 3x3 depthwise conv, padding=1 (cross-correlation, matches lax.conv).
// One thread per output element; x-contiguous -> coalesced streaming.
// ---------------------------------------------------------------------------
__global__ void dwconv3x3(const float* __restrict__ in, const float* __restrict__ w,
                          float* __restrict__ out) {
  size_t idx = (size_t)blockIdx.x * 256 + threadIdx.x;   // [b*576, y, x]
  int x  = (int)(idx & 127);
  int y  = (int)((idx >> 7) & 127);
  size_t plane = idx >> 14;                // b*576 + ch
  int ch = (int)(plane % QKV_CH);
  const float* wp = w + (size_t)ch * 9;
  const float* p  = in + (plane << 14);

  float acc = 0.f;
#pragma unroll
  for (int ky = 0; ky < 3; ++ky) {
    int yy = y + ky - 1;
    if (yy < 0 || yy > 127) continue;
#pragma unroll
    for (int kx = 0; kx < 3; ++kx) {
      int xx = x + kx - 1;
      if (xx < 0 || xx > 127) continue;
      acc += wp[ky * 3 + kx] * p[yy * 128 + xx];
    }
  }
  out[idx] = acc;
}

// ---------------------------------------------------------------------------
// Per-row inverse L2 norm over the spatial axis for q (sel=0) and k (sel=1).
// Row id = ((b*2 + sel)*192 + ch). One 256-thread block per row.
// ---------------------------------------------------------------------------
__global__ void rownorm(const float* __restrict__ dwq, float* __restrict__ invn) {
  int row = blockIdx.x;
  int b   = row / (2 * DIM);
  int rem = row % (2 * DIM);
  int sel = rem / DIM;       // 0=q, 1=k
  int ch  = rem % DIM;
  const float* p = dwq + ((size_t)b * QKV_CH + sel * DIM + ch) * GN;

  float s = 0.f;
  for (int i = threadIdx.x; i < GN; i += 256) {
    float v = p[i];
    s += v * v;
  }
  __shared__ float red[256];
  red[threadIdx.x] = s;
  __syncthreads();
  for (int off = 128; off > 0; off >>= 1) {
    if (threadIdx.x < off) red[threadIdx.x] += red[threadIdx.x + off];
    __syncthreads();
  }
  if (threadIdx.x == 0)
    invn[row] = 1.f / fmaxf(sqrtf(red[0]), 1e-12f);
}

// ---------------------------------------------------------------------------
// Partial Gram matrices: part[b,h,s][c,d] = sum_{n in split s} q[c,n]*k[d,n].
// Grid (HEADS, 8, NSPLIT) -> 1024 blocks (parallelism for the big chip).
// 128 threads = 4 waves; waves 0..2 own 16-row strips (3 accumulators over
// the 48 d-columns); all threads cooperate on padded-LDS staging.
// Deterministic: partials written to ws, reduced in fixed order (no atomics).
// ---------------------------------------------------------------------------
__global__ void attn_partial(const float* __restrict__ dwq, float* __restrict__ part) {
  const int h = blockIdx.x, b = blockIdx.y, s = blockIdx.z;
  __shared__ float Qs[HDIM][65];
  __shared__ float Ks[HDIM][65];

  const int tid  = threadIdx.x;
  const int lane = tid & 31;
  const int wave = tid >> 5;
  const int l16  = lane & 15;
  const int kk   = (lane < 16) ? 0 : 2;

  const float* qb = dwq + ((size_t)b * QKV_CH + h * HDIM) * GN;
  const float* kb = dwq + ((size_t)b * QKV_CH + DIM + h * HDIM) * GN;
  const int nbase = s * (GN / NSPLIT);      // 512 columns per split

  v8f acc0 = {}, acc1 = {}, acc2 = {};

  for (int nc = 0; nc < GN / NSPLIT; nc += 64) {
    const int n0 = nbase + nc;
    __syncthreads();
    for (int i = tid; i < HDIM * 64; i += 128) {
      int r = i >> 6, c = i & 63;
      Qs[r][c] = qb[(size_t)r * GN + n0 + c];
      Ks[r][c] = kb[(size_t)r * GN + n0 + c];
    }
    __syncthreads();

    if (wave < 3) {
      for (int k = 0; k < 64; k += 4) {
        v2f a;
        a.x = Qs[wave * 16 + l16][k + kk];
        a.y = Qs[wave * 16 + l16][k + kk + 1];
        v2f b0; b0.x = Ks[l16][k + kk];      b0.y = Ks[l16][k + kk + 1];
        v2f b1; b1.x = Ks[16 + l16][k + kk]; b1.y = Ks[16 + l16][k + kk + 1];
        v2f b2; b2.x = Ks[32 + l16][k + kk]; b2.y = Ks[32 + l16][k + kk + 1];
        acc0 = __builtin_amdgcn_wmma_f32_16x16x4_f32(false, a, false, b0, (short)0, acc0, false, false);
        acc1 = __builtin_amdgcn_wmma_f32_16x16x4_f32(false, a, false, b1, (short)0, acc1, false, false);
        acc2 = __builtin_amdgcn_wmma_f32_16x16x4_f32(false, a, false, b2, (short)0, acc2, false, false);
      }
    }
  }

  if (wave < 3) {
    float* pp = part + (((size_t)(b * HEADS + h) * NSPLIT + s) * HDIM) * HDIM;
#pragma unroll
    for (int r = 0; r < 8; ++r) {
      int c = wave * 16 + r + ((lane < 16) ? 0 : 8);
      pp[c * HDIM + l16]      = acc0[r];
      pp[c * HDIM + 16 + l16] = acc1[r];
      pp[c * HDIM + 32 + l16] = acc2[r];
    }
  }
}

// ---------------------------------------------------------------------------
// Reduce NSPLIT partials (fixed order), scale by inv-norms * temperature,
// row-softmax over d, write attn[b,h,48,48]. Grid (HEADS, 8), block 256.
// ---------------------------------------------------------------------------
__global__ void attn_reduce_softmax(const float* __restrict__ part,
                                    const float* __restrict__ invn,
                                    const float* __restrict__ temp,
                                    float* __restrict__ attn) {
  const int h = blockIdx.x, b = blockIdx.y;
  __shared__ float S[HDIM][49];
  const int tid = threadIdx.x;

  const float* pp  = part + (size_t)(b * HEADS + h) * NSPLIT * HDIM * HDIM;
  const float* ivq = invn + (size_t)b * 2 * DIM + h * HDIM;
  const float* ivk = invn + (size_t)b * 2 * DIM + DIM + h * HDIM;
  const float  tm  = temp[h];

  for (int i = tid; i < HDIM * HDIM; i += 256) {
    float s = 0.f;
    for (int sp = 0; sp < NSPLIT; ++sp) s += pp[(size_t)sp * HDIM * HDIM + i];
    int c = i / HDIM, d = i - c * HDIM;
    S[c][d] = s * ivq[c] * ivk[d] * tm;
  }
  __syncthreads();

  if (tid < HDIM) {
    float mx = -INFINITY;
    for (int d = 0; d < HDIM; ++d) mx = fmaxf(mx, S[tid][d]);
    float sum = 0.f;
    for (int d = 0; d < HDIM; ++d) {
      float e = __expf(S[tid][d] - mx);
      S[tid][d] = e;
      sum += e;
    }
    float inv = 1.f / sum;
    float* ap = attn + (((size_t)b * HEADS + h) * HDIM + tid) * HDIM;
    for (int d = 0; d < HDIM; ++d) ap[d] = S[tid][d] * inv;
  }
}

// ---------------------------------------------------------------------------
// Fold projection into attention:  Mb[b][o, h*48+d] = sum_c proj_w[o,h*48+c]*attn[b,h,c,d]
// Then out = Mb @ v is a single GEMM (saves the 100MB attn*v intermediate).
// ---------------------------------------------------------------------------
__global__ void combine(const float* __restrict__ proj_w, const float* __restrict__ attn,
                        float* __restrict__ Mb) {
  int o = blockIdx.x % DIM;
  int b = blockIdx.x / DIM;
  int d = threadIdx.x;                // 0..191
  int hh = d / HDIM, dd = d % HDIM;
  const float* pw = proj_w + (size_t)o * DIM + hh * HDIM;
  const float* at = attn + (((size_t)b * HEADS + hh) * HDIM) * HDIM + dd;
  float s = 0.f;
  for (int c = 0; c < HDIM; ++c) s += pw[c] * at[(size_t)c * HDIM];
  Mb[((size_t)b * DIM + o) * DIM + d] = s;
}

// ---------------------------------------------------------------------------
extern "C" void kernel_launch(void* const* d_in, const int* in_sizes, int n_in,
                              void* d_out, int out_size, void* d_ws, size_t ws_size,
                              hipStream_t stream) {
  const float* x      = (const float*)d_in[0];   // [8,192,128,128]
  const float* qkv_w  = (const float*)d_in[1];   // [576,192]
  const float* dw_w   = (const float*)d_in[2];   // [576,1,3,3]
  const float* proj_w = (const float*)d_in[3];   // [192,192]
  const float* temperature = (const float*)d_in[4]; // [4]
  float* out = (float*)d_out;                    // [8,192,128,128]

  const size_t planeElems = (size_t)8 * QKV_CH * GN;   // 75,497,472
  float* qkv  = (float*)d_ws;                          // qkv pre-dwconv
  float* dwq  = qkv + planeElems;                      // qkv post-dwconv
  float* invn = dwq + planeElems;                      // [8,2,192]
  float* part = invn + (size_t)8 * 2 * DIM;            // [8,4,NSPLIT,48,48]
  float* attn = part + (size_t)8 * HEADS * NSPLIT * HDIM * HDIM;
  float* Mb   = attn + (size_t)8 * HEADS * HDIM * HDIM;// [8,192,192]

  // 1) qkv = qkv_w @ x   (shared weights, per-batch X)
  gemm_wmma<<<dim3(QKV_CH / 16, GN / 512, 8), 256, 0, stream>>>(
      qkv_w, 0, x, (long)DIM * GN, 0, qkv, QKV_CH, DIM);

  // 2) 3x3 depthwise conv
  dwconv3x3<<<(unsigned)(planeElems / 256), 256, 0, stream>>>(qkv, dw_w, dwq);

  // 3) inverse L2 norms for q and k rows
  rownorm<<<8 * 2 * DIM, 256, 0, stream>>>(dwq, invn);

  // 4) partial 48x48 Gram tiles, 1024 blocks
  attn_partial<<<dim3(HEADS, 8, NSPLIT), 128, 0, stream>>>(dwq, part);

  // 5) deterministic reduce + scale + softmax
  attn_reduce_softmax<<<dim3(HEADS, 8), 256, 0, stream>>>(part, invn, temperature, attn);

  // 6) fold proj_w through attn -> per-batch [192,192]
  combine<<<8 * DIM, DIM, 0, stream>>>(proj_w, attn, Mb);

  // 7) out = Mb @ v   (per-batch weights, v = dwq channels [384,576))
  gemm_wmma<<<dim3(DIM / 16, GN / 512, 8), 256, 0, stream>>>(
      Mb, (long)DIM * DIM, dwq, (long)QKV_CH * GN, 2 * DIM, out, DIM, DIM);
}